// GCN_no_pooling_34273839022398
// MI455X (gfx1250) — compile-verified
//
#include <hip/hip_runtime.h>
#include <hip/hip_bf16.h>

typedef __attribute__((ext_vector_type(2))) float v2f;
typedef __attribute__((ext_vector_type(8))) float v8f;

#define N_NODES 50000
#define FEAT 128

// ---------------------------------------------------------------------------
// fp32 global atomic add (hardware global_atomic_add_f32, never a CAS loop)
// ---------------------------------------------------------------------------
__device__ __forceinline__ void atomicAddF32(float* p, float v) {
#if __has_builtin(__builtin_amdgcn_global_atomic_fadd_f32)
  __builtin_amdgcn_global_atomic_fadd_f32(
      (__attribute__((address_space(1))) float*)p, v);
#else
  __hip_atomic_fetch_add(p, v, __ATOMIC_RELAXED, __HIP_MEMORY_SCOPE_AGENT);
#endif
}

// ---------------------------------------------------------------------------
// small utility kernels
// ---------------------------------------------------------------------------
__global__ void fill_zero_kernel(float* __restrict__ p, int n) {
  int i = blockIdx.x * blockDim.x + threadIdx.x;
  if (i < n) p[i] = 0.0f;
}

// deg[col[e]] += 1   (in-degree of targets)
__global__ void deg_count_kernel(const long long* __restrict__ ei,
                                 float* __restrict__ deg, int E) {
  int e = blockIdx.x * blockDim.x + threadIdx.x;
  if (e < E) {
    long long c = ei[(long long)E + e];
    atomicAddF32(&deg[c], 1.0f);
  }
}

// dinv = deg>0 ? rsqrt(deg) : 0   (in place)
__global__ void dinv_kernel(float* __restrict__ deg, int n) {
  int i = blockIdx.x * blockDim.x + threadIdx.x;
  if (i < n) {
    float d = deg[i];
    deg[i] = (d > 0.0f) ? rsqrtf(d) : 0.0f;
  }
}

// out[i*FEAT + f] = b[f]  (pre-load aggregation buffer with bias)
__global__ void init_bias_rows_kernel(float* __restrict__ out,
                                      const float* __restrict__ b, int total) {
  int i = blockIdx.x * blockDim.x + threadIdx.x;
  if (i < total) out[i] = b[i & (FEAT - 1)];
}

// ---------------------------------------------------------------------------
// edge scatter: out[col] += h[row] * dinv[row]*dinv[col]
// 32 lanes per edge, float4 per lane: 1x global_load_b128 + 4x f32 atomics.
// 8 edges per 256-thread block.
// ---------------------------------------------------------------------------
__global__ void scatter_edges_kernel(const float* __restrict__ h,
                                     const long long* __restrict__ ei,
                                     const float* __restrict__ dinv,
                                     float* __restrict__ out, int E) {
  int e = blockIdx.x * 8 + (threadIdx.x >> 5);
  if (e >= E) return;
  int f4 = (threadIdx.x & 31) * 4;  // 32 lanes * 4 floats = 128 features
  long long row = ei[e];
  long long col = ei[(long long)E + e];
  float norm = dinv[row] * dinv[col];
  const float4 hv = *(const float4*)&h[row * FEAT + f4];
  float* o = &out[col * FEAT + f4];
  atomicAddF32(o + 0, hv.x * norm);
  atomicAddF32(o + 1, hv.y * norm);
  atomicAddF32(o + 2, hv.z * norm);
  atomicAddF32(o + 3, hv.w * norm);
}

// ---------------------------------------------------------------------------
// fp32 WMMA GEMM:  C[M x NOUT] = op(A)[M x 128] * W[128 x NOUT] (+bias, relu)
// One block = 16-row strip of A staged in LDS (padded stride 132 -> conflict-
// free 16x4 fragment reads). One wave per 16-col tile: blockDim = 32*(NOUT/16).
// NOUT is a template constant so all B loads / C stores use immediate offsets
// from a single per-lane base (no per-iteration 64-bit address math).
// ---------------------------------------------------------------------------
template <int NOUT, bool RELU_IN, bool BIAS_RELU>
__global__ void gemm_wmma_k128_kernel(const float* __restrict__ A,
                                      const float* __restrict__ W,
                                      const float* __restrict__ bias,
                                      float* __restrict__ C) {
  __shared__ float As[16][132];

  const int row0 = blockIdx.x * 16;

  // cooperative load of 16x128 A-strip (optionally fused ReLU on input)
  for (int i = threadIdx.x; i < 16 * FEAT; i += blockDim.x) {
    int r = i >> 7;
    int c = i & (FEAT - 1);
    float v = A[(long long)(row0 + r) * FEAT + c];
    if (RELU_IN) v = fmaxf(v, 0.0f);
    As[r][c] = v;
  }
  __syncthreads();

  const int lane = threadIdx.x & 31;
  const int wave = threadIdx.x >> 5;
  const int n0 = wave * 16;
  const int m = lane & 15;         // A row / B,C column within tile
  const int kh = (lane >> 4) * 2;  // K sub-pair: 0 for lanes 0-15, 2 else

  // per-lane B base: column n0+m, row kh; rows advance by NOUT floats
  const float* __restrict__ Wp = W + (long long)kh * NOUT + (n0 + m);

  v8f acc = {0.f, 0.f, 0.f, 0.f, 0.f, 0.f, 0.f, 0.f};

#pragma unroll
  for (int k = 0; k < FEAT; k += 4) {
    v2f a, b;
    // A fragment (16x4 fp32): lanes 0-15 -> K=k,k+1 ; lanes 16-31 -> K=k+2,k+3
    a.x = As[m][k + kh];
    a.y = As[m][k + kh + 1];
    // B fragment (4x16 fp32): same K split; immediate offsets (k*NOUT const)
    b.x = Wp[(long long)k * NOUT];
    b.y = Wp[(long long)(k + 1) * NOUT];
    acc = __builtin_amdgcn_wmma_f32_16x16x4_f32(false, a, false, b, (short)0,
                                                acc, false, false);
  }

  // D layout: VGPR r -> row r (lanes 0-15) / row r+8 (lanes 16-31), col=lane%16
  const int mbase = row0 + ((lane >> 4) << 3);
  const int col = n0 + m;
  float bv = 0.0f;
  if (BIAS_RELU) bv = bias[col];
  float* __restrict__ Cp = C + (long long)mbase * NOUT + col;
#pragma unroll
  for (int r = 0; r < 8; ++r) {
    float v = acc[r];
    if (BIAS_RELU) v = fmaxf(v + bv, 0.0f);
    Cp[(long long)r * NOUT] = v;
  }
}

// ---------------------------------------------------------------------------
// launcher
// ---------------------------------------------------------------------------
extern "C" void kernel_launch(void* const* d_in, const int* in_sizes, int n_in,
                              void* d_out, int out_size, void* d_ws,
                              size_t ws_size, hipStream_t stream) {
  const float* x      = (const float*)d_in[0];
  const long long* ei = (const long long*)d_in[1];
  const float* W1     = (const float*)d_in[2];
  const float* b1     = (const float*)d_in[3];
  const float* W2     = (const float*)d_in[4];
  const float* b2     = (const float*)d_in[5];
  const float* Wf1    = (const float*)d_in[6];
  const float* bf1    = (const float*)d_in[7];
  const float* Wf2    = (const float*)d_in[8];
  const float* bf2    = (const float*)d_in[9];
  float* out          = (float*)d_out;

  const int N = N_NODES;
  const int E = in_sizes[1] / 2;  // 800000
  const int NF = N * FEAT;

  // workspace partition (256B aligned)
  char* ws = (char*)d_ws;
  float* dinv = (float*)ws;                         // N floats
  size_t off = ((size_t)N * 4 + 255) & ~(size_t)255;
  float* bufA = (float*)(ws + off);                 // N*128 floats
  float* bufB = bufA + (size_t)NF;                  // N*128 floats

  const int T = 256;
  const int nBlocksN  = (N + T - 1) / T;
  const int nBlocksE  = (E + T - 1) / T;
  const int nBlocksNF = (NF + T - 1) / T;
  const int nBlocksSc = (E + 7) / 8;
  const int rowStrips = N / 16;                     // 3125 exact

  // 1) degree -> dinv
  fill_zero_kernel<<<nBlocksN, T, 0, stream>>>(dinv, N);
  deg_count_kernel<<<nBlocksE, T, 0, stream>>>(ei, dinv, E);
  dinv_kernel<<<nBlocksN, T, 0, stream>>>(dinv, N);

  // 2) conv1: h1 = x @ W1 ; agg1 = scatter(h1) + b1 (ReLU fused into next A-load)
  gemm_wmma_k128_kernel<128, false, false>
      <<<rowStrips, 256, 0, stream>>>(x, W1, nullptr, bufA);
  init_bias_rows_kernel<<<nBlocksNF, T, 0, stream>>>(bufB, b1, NF);
  scatter_edges_kernel<<<nBlocksSc, 256, 0, stream>>>(bufA, ei, dinv, bufB, E);

  // 3) conv2: h2 = relu(agg1) @ W2 ; agg2 = scatter(h2) + b2
  gemm_wmma_k128_kernel<128, true, false>
      <<<rowStrips, 256, 0, stream>>>(bufB, W2, nullptr, bufA);
  init_bias_rows_kernel<<<nBlocksNF, T, 0, stream>>>(bufB, b2, NF);
  scatter_edges_kernel<<<nBlocksSc, 256, 0, stream>>>(bufA, ei, dinv, bufB, E);

  // 4) dense: h3 = relu(relu(agg2) @ Wf1 + bf1)
  gemm_wmma_k128_kernel<128, true, true>
      <<<rowStrips, 256, 0, stream>>>(bufB, Wf1, bf1, bufA);
  // 5) dense: out = relu(h3 @ Wf2 + bf2)   (NOUT = 64 -> 4 waves/block)
  gemm_wmma_k128_kernel<64, false, true>
      <<<rowStrips, 128, 0, stream>>>(bufA, Wf2, bf2, out);
}